// MaxPool3d_22376779612829
// MI455X (gfx1250) — compile-verified
//
#include <hip/hip_runtime.h>
#include <stdint.h>

#ifndef __has_builtin
#define __has_builtin(x) 0
#endif
#if __has_builtin(__builtin_amdgcn_tensor_load_to_lds) && __has_builtin(__builtin_amdgcn_s_wait_tensorcnt)
#define HAVE_TDM 1
#else
#define HAVE_TDM 0
#endif

typedef __attribute__((ext_vector_type(4))) unsigned int v4u;
typedef __attribute__((ext_vector_type(4))) int         v4i;
typedef __attribute__((ext_vector_type(8))) int         v8i;

#define NBITS            (1u << 27)                     // compact-key capacity (bits)
#define NWORDS           (1u << 22)                     // bitmap words (16 MB)
#define SCAN_BLOCK       256
#define SCAN_WPT         8
#define WORDS_PER_SCANBLK (SCAN_BLOCK * SCAN_WPT)       // 2048
#define NUM_SCANBLKS     (NWORDS / WORDS_PER_SCANBLK)   // 2048
#define ENC_BLOCKS       2048                           // persistent encode blocks

__device__ __forceinline__ uint32_t map_f32(float f) {
  uint32_t b = __float_as_uint(f);
  return (b & 0x80000000u) ? ~b : (b | 0x80000000u);   // order-preserving float->uint
}
__device__ __forceinline__ float unmap_f32(uint32_t u) {
  uint32_t b = (u & 0x80000000u) ? (u ^ 0x80000000u) : ~u;
  return __uint_as_float(b);
}

__global__ void k_zero_u32(uint32_t* __restrict__ p, long long n) {
  long long i = (long long)blockIdx.x * blockDim.x + threadIdx.x;
  long long stride = (long long)gridDim.x * blockDim.x;
  for (; i < n; i += stride) p[i] = 0u;
}

__global__ void k_init_minmax(int* __restrict__ minmax) {
  int t = threadIdx.x;
  if (t < 4) minmax[t] = 0x7FFFFFFF;
  else if (t < 8) minmax[t] = (int)0x80000000;
}

__global__ void k_minmax(const int* __restrict__ coords, int* __restrict__ minmax, long long L) {
  int mn[4] = {0x7FFFFFFF, 0x7FFFFFFF, 0x7FFFFFFF, 0x7FFFFFFF};
  int mx[4] = {(int)0x80000000, (int)0x80000000, (int)0x80000000, (int)0x80000000};
  long long stride = (long long)gridDim.x * blockDim.x;
  for (long long i = (long long)blockIdx.x * blockDim.x + threadIdx.x; i < L; i += stride) {
    int4 c = ((const int4*)coords)[i];
    int v[4] = {c.x, c.y, c.z, c.w};
#pragma unroll
    for (int d = 0; d < 4; ++d) { mn[d] = min(mn[d], v[d]); mx[d] = max(mx[d], v[d]); }
  }
#pragma unroll
  for (int o = 16; o > 0; o >>= 1) {
#pragma unroll
    for (int d = 0; d < 4; ++d) {
      mn[d] = min(mn[d], __shfl_down(mn[d], o, 32));
      mx[d] = max(mx[d], __shfl_down(mx[d], o, 32));
    }
  }
  if ((threadIdx.x & 31) == 0) {
#pragma unroll
    for (int d = 0; d < 4; ++d) { atomicMin(&minmax[d], mn[d]); atomicMax(&minmax[4 + d], mx[d]); }
  }
}

// Encode both the reference Horner key (original sizes) and an order-isomorphic
// compact key (< 2^27) used for bitmap ranking. Persistent blocks stream coord
// tiles through a double-buffered TDM (tensor DMA) pipeline: wave 0 issues the
// DMA for the NEXT tile, then s_wait_tensorcnt(1) guarantees (in-order per-wave
// TENSORcnt) that the CURRENT tile has landed while the next one is in flight.
__global__ void k_encode(const int* __restrict__ coords, const int* __restrict__ ksp,
                         const int* __restrict__ minmax, int* __restrict__ enc,
                         uint32_t* __restrict__ cenc, uint32_t* __restrict__ bitmap,
                         long long L, int numTiles) {
  int ks = ksp[0];
  int mn[4], sz[4];
#pragma unroll
  for (int d = 0; d < 4; ++d) { mn[d] = minmax[d]; sz[d] = minmax[4 + d] - mn[d] + 1; }

#if HAVE_TDM
  __shared__ __align__(16) int s_tile[2][256 * 4];     // sole LDS var -> offset 0 / 4096

  auto issue = [&](int tile, int buf) {
    uint64_t ga = (uint64_t)(uintptr_t)(coords + (long long)tile * 1024);
    long long remain = L * 4 - (long long)tile * 1024;
    if (remain > 1024) remain = 1024;
    uint32_t td0 = (uint32_t)remain;                   // tensor_dim0: OOB reads -> 0
    v4u g0;
    g0.x = 1u;                                         // count=1, load descriptor
    g0.y = buf ? 4096u : 0u;                           // lds_addr
    g0.z = (uint32_t)ga;                               // global_addr[31:0]
    g0.w = (uint32_t)((ga >> 32) & 0x01FFFFFFull) | 0x80000000u; // addr hi | type=2
    v8i g1;
    g1[0] = (int)(2u << 16);                           // data_size = 4B
    g1[1] = (int)((td0 & 0xFFFFu) << 16);              // tensor_dim0[15:0] @ bits 63:48
    g1[2] = (int)((td0 >> 16) & 0xFFFFu);              // tensor_dim0[31:16]; tensor_dim1=0
    g1[3] = (int)(1024u << 16);                        // tile_dim0 = 1024 elements
    g1[4] = 0; g1[5] = 0; g1[6] = 0; g1[7] = 0;        // tile_dim1/2=0, strides=0
    v4i z4 = (v4i)0;
#if __clang_major__ >= 23
    v8i z8 = (v8i)0;
    __builtin_amdgcn_tensor_load_to_lds(g0, g1, z4, z4, z8, 0);
#else
    __builtin_amdgcn_tensor_load_to_lds(g0, g1, z4, z4, 0);
#endif
  };

  int tile = blockIdx.x;
  if (threadIdx.x < 32 && tile < numTiles) issue(tile, 0);   // prologue: tile -> buf0
  int it = 0;
  for (; tile < numTiles; tile += gridDim.x, ++it) {
    int cur = it & 1;
    int nxt = tile + gridDim.x;
    if (threadIdx.x < 32) {                            // wave 0 runs the pipeline
      if (nxt < numTiles) {
        issue(nxt, cur ^ 1);                           // overlap DMA of next tile
        __builtin_amdgcn_s_wait_tensorcnt(1);          // current tile complete
      } else {
        __builtin_amdgcn_s_wait_tensorcnt(0);          // drain
      }
    }
    __syncthreads();                                   // tile visible to all waves
    long long i = (long long)tile * 256 + threadIdx.x;
    if (i < L) {
      int4 c = ((const int4*)s_tile[cur])[threadIdx.x];
#else
  {
    long long stride = (long long)gridDim.x * blockDim.x;
    for (long long i = (long long)blockIdx.x * blockDim.x + threadIdx.x; i < L; i += stride) {
      int4 c = ((const int4*)coords)[i];
#endif
      int cc[4] = {c.x, c.y, c.z, c.w};
      int oc[4];
      oc[0] = cc[0];
#pragma unroll
      for (int d = 1; d < 4; ++d) oc[d] = cc[d] / ks;  // coords are non-negative
      long long e = 0;
#pragma unroll
      for (int d = 0; d < 4; ++d) e = e * sz[d] + (oc[d] - mn[d]);
      enc[i] = (int)e;                                 // < 2^31 per reference
      // compact key: same lexicographic order, digits shifted to [0, csz)
      uint64_t ce = (uint32_t)(oc[0] - mn[0]);
#pragma unroll
      for (int d = 1; d < 4; ++d) {
        int cmn = mn[d] / ks;
        int csz = minmax[4 + d] / ks - cmn + 1;
        ce = ce * (uint32_t)csz + (uint32_t)(oc[d] - cmn);
      }
      uint32_t ce32 = (ce < (uint64_t)NBITS) ? (uint32_t)ce : (NBITS - 1u);
      cenc[i] = ce32;
      atomicOr(&bitmap[ce32 >> 5], 1u << (ce32 & 31u));
    }
#if HAVE_TDM
    __syncthreads();                                   // buffer free before reuse
#endif
  }
}

// Per-word popcount prefix within each 2048-word tile + per-tile sums.
__global__ void k_scan1(const uint32_t* __restrict__ bitmap, uint32_t* __restrict__ prefix,
                        uint32_t* __restrict__ blkSums) {
  __shared__ uint32_t sdata[SCAN_BLOCK];
  int tid = threadIdx.x;
  long long base = (long long)blockIdx.x * WORDS_PER_SCANBLK + (long long)tid * SCAN_WPT;
  const uint4* bm4 = (const uint4*)bitmap;
  uint4 a = bm4[base >> 2];
  uint4 b = bm4[(base >> 2) + 1];
  uint32_t w[8] = {a.x, a.y, a.z, a.w, b.x, b.y, b.z, b.w};
  uint32_t run[8], s = 0;
#pragma unroll
  for (int k = 0; k < 8; ++k) { run[k] = s; s += (uint32_t)__popc(w[k]); }
  sdata[tid] = s;
  __syncthreads();
  for (int off = 1; off < SCAN_BLOCK; off <<= 1) {
    uint32_t v = (tid >= off) ? sdata[tid - off] : 0u;
    __syncthreads();
    sdata[tid] += v;
    __syncthreads();
  }
  uint32_t excl = sdata[tid] - s;
#pragma unroll
  for (int k = 0; k < 8; ++k) prefix[base + k] = excl + run[k];
  if (tid == SCAN_BLOCK - 1) blkSums[blockIdx.x] = sdata[tid];
}

// Exclusive scan of the 2048 tile sums (single block), total -> *total.
__global__ void k_scan2(uint32_t* __restrict__ blkSums, uint32_t* __restrict__ total) {
  __shared__ uint32_t sdata[SCAN_BLOCK];
  int tid = threadIdx.x;
  int base = tid * SCAN_WPT;
  uint32_t w[8], run[8], s = 0;
#pragma unroll
  for (int k = 0; k < 8; ++k) w[k] = blkSums[base + k];
#pragma unroll
  for (int k = 0; k < 8; ++k) { run[k] = s; s += w[k]; }
  sdata[tid] = s;
  __syncthreads();
  for (int off = 1; off < SCAN_BLOCK; off <<= 1) {
    uint32_t v = (tid >= off) ? sdata[tid - off] : 0u;
    __syncthreads();
    sdata[tid] += v;
    __syncthreads();
  }
  uint32_t excl = sdata[tid] - s;
#pragma unroll
  for (int k = 0; k < 8; ++k) blkSums[base + k] = excl + run[k];
  if (tid == SCAN_BLOCK - 1) *total = sdata[tid];
}

// One wave32 per point: lane 0 computes the sorted-unique rank; all 32 lanes
// scatter-max one channel each with native u32 atomics (monotone float map).
__global__ void k_scatter(const float* __restrict__ feats, const uint32_t* __restrict__ cenc,
                          const int* __restrict__ enc, const uint32_t* __restrict__ bitmap,
                          const uint32_t* __restrict__ prefix, const uint32_t* __restrict__ blkSums,
                          uint32_t* __restrict__ outU, int* __restrict__ keyAtRank, long long L) {
  long long t = (long long)blockIdx.x * blockDim.x + threadIdx.x;
  long long p = t >> 5;
  int lane = threadIdx.x & 31;
  if (p >= L) return;
  __builtin_prefetch(feats + (p + 64) * 32, 0, 0);  // global_prefetch_b8
  uint32_t r = 0;
  if (lane == 0) {
    uint32_t ce = cenc[p];
    uint32_t w = ce >> 5, bit = ce & 31u;
    uint32_t word = bitmap[w];
    r = blkSums[w >> 11] + prefix[w] + (uint32_t)__popc(word & ((1u << bit) - 1u));
    keyAtRank[r] = enc[p];   // all members of a group write the same value
  }
  r = __builtin_amdgcn_readfirstlane(r);
  float f = feats[p * 32 + lane];
  atomicMax(&outU[(long long)r * 32 + lane], map_f32(f));
}

__global__ void k_finalize(uint32_t* __restrict__ outU, const uint32_t* __restrict__ total,
                           long long n) {
  long long i = (long long)blockIdx.x * blockDim.x + threadIdx.x;
  if (i >= n) return;
  uint32_t nu = *total;
  uint32_t u = outU[i];
  float f = ((uint32_t)(i >> 5) < nu) ? unmap_f32(u) : 0.0f;
  ((float*)outU)[i] = f;   // in-place uint->float, same slot, same thread
}

__global__ void k_decode(const int* __restrict__ keyAtRank, const uint32_t* __restrict__ total,
                         const int* __restrict__ minmax, float* __restrict__ out, long long L) {
  long long r = (long long)blockIdx.x * blockDim.x + threadIdx.x;
  if (r >= L) return;
  uint32_t nu = *total;
  float* dec = out + L * 32;
  float* val = out + L * 36;
  if ((uint64_t)r < (uint64_t)nu) {
    int cur = keyAtRank[r];
    int d[4];
#pragma unroll
    for (int idx = 3; idx >= 0; --idx) {
      int mnv = minmax[idx], szv = minmax[4 + idx] - mnv + 1;
      d[idx] = mnv + cur % szv;
      cur /= szv;
    }
    dec[r * 4 + 0] = (float)d[0];
    dec[r * 4 + 1] = (float)d[1];
    dec[r * 4 + 2] = (float)d[2];
    dec[r * 4 + 3] = (float)d[3];
    val[r] = 1.0f;
  } else {
    dec[r * 4 + 0] = 0.0f; dec[r * 4 + 1] = 0.0f;
    dec[r * 4 + 2] = 0.0f; dec[r * 4 + 3] = 0.0f;
    val[r] = 0.0f;
  }
}

extern "C" void kernel_launch(void* const* d_in, const int* in_sizes, int n_in,
                              void* d_out, int out_size, void* d_ws, size_t ws_size,
                              hipStream_t stream) {
  const float* feats = (const float*)d_in[0];
  const int* coords = (const int*)d_in[1];
  const int* ksp = (const int*)d_in[2];
  long long L = (long long)in_sizes[1] / 4;
  (void)n_in; (void)out_size; (void)ws_size;

  char* ws = (char*)d_ws;
  size_t off = 0;
  auto alloc = [&](size_t bytes) -> void* {
    void* p = ws + off;
    off = (off + bytes + 255) & ~(size_t)255;
    return p;
  };
  uint32_t* bitmap    = (uint32_t*)alloc((size_t)NWORDS * 4);        // 16 MB
  uint32_t* prefix    = (uint32_t*)alloc((size_t)NWORDS * 4);        // 16 MB
  uint32_t* blkSums   = (uint32_t*)alloc((size_t)NUM_SCANBLKS * 4);
  uint32_t* total     = (uint32_t*)alloc(4);
  int*      minmax    = (int*)alloc(8 * 4);
  int*      enc       = (int*)alloc((size_t)L * 4);                  // 8 MB
  uint32_t* cenc      = (uint32_t*)alloc((size_t)L * 4);             // 8 MB
  int*      keyAtRank = (int*)alloc((size_t)L * 4);                  // 8 MB

  uint32_t* outU = (uint32_t*)d_out;

  k_zero_u32<<<2048, 256, 0, stream>>>(bitmap, (long long)NWORDS);
  k_zero_u32<<<4096, 256, 0, stream>>>(outU, L * 32);
  k_init_minmax<<<1, 32, 0, stream>>>(minmax);
  k_minmax<<<1024, 256, 0, stream>>>(coords, minmax, L);
  int numTiles = (int)((L + 255) / 256);
  k_encode<<<ENC_BLOCKS, 256, 0, stream>>>(coords, ksp, minmax, enc, cenc, bitmap, L, numTiles);
  k_scan1<<<NUM_SCANBLKS, SCAN_BLOCK, 0, stream>>>(bitmap, prefix, blkSums);
  k_scan2<<<1, SCAN_BLOCK, 0, stream>>>(blkSums, total);
  int scatBlocks = (int)((L * 32 + 255) / 256);
  k_scatter<<<scatBlocks, 256, 0, stream>>>(feats, cenc, enc, bitmap, prefix, blkSums,
                                            outU, keyAtRank, L);
  k_finalize<<<scatBlocks, 256, 0, stream>>>(outU, total, L * 32);
  int rowBlocks = (int)((L + 255) / 256);
  k_decode<<<rowBlocks, 256, 0, stream>>>(keyAtRank, total, minmax, (float*)d_out, L);
}